// GCN_7000796692945
// MI455X (gfx1250) — compile-verified
//
#include <hip/hip_runtime.h>
#include <hip/hip_bf16.h>

typedef __attribute__((ext_vector_type(2))) float v2f;
typedef __attribute__((ext_vector_type(8))) float v8f;

// ---------------------------------------------------------------------------
// Hardware fp32 atomic add (global_atomic_add_f32), avoids CAS loop.
// ---------------------------------------------------------------------------
__device__ __forceinline__ void atomAddF(float* p, float v) {
    unsafeAtomicAdd(p, v);
}

// ---------------------------------------------------------------------------
// Generic fill
// ---------------------------------------------------------------------------
__global__ void fill_kernel(float* __restrict__ p, float v, long long n) {
    long long i = (long long)blockIdx.x * blockDim.x + threadIdx.x;
    if (i < n) p[i] = v;
}

// ---------------------------------------------------------------------------
// Degree accumulation: deg[dst[e]] += 1  (deg pre-filled with 1.0 self-loop)
// ---------------------------------------------------------------------------
__global__ void degree_kernel(const int* __restrict__ dst, float* __restrict__ deg, int E) {
    int e = blockIdx.x * blockDim.x + threadIdx.x;
    if (e < E) atomAddF(&deg[dst[e]], 1.0f);
}

__global__ void rsqrt_kernel(float* __restrict__ deg, int n) {
    int i = blockIdx.x * blockDim.x + threadIdx.x;
    if (i < n) deg[i] = rsqrtf(deg[i]);
}

// ---------------------------------------------------------------------------
// WMMA fp32 GEMM: C[M,Ncols] = A[M,K] @ B[K,Ncols], row-major.
//
// Block = 256 threads = 8 waves. The whole weight matrix B is staged into LDS
// once per block, zero-padded to NP = CT*16 columns (so no bounds masks on B
// and no masks needed for the OUT=40 -> 48 padding). Each wave then computes a
// 16 x NP stripe of C: the A fragment is loaded ONCE per k-step and reused
// across CT WMMAs (acc[CT]), so A is read exactly once from HBM.
//
// Fragment layouts per CDNA5 ISA 7.12.2 (V_WMMA_F32_16X16X4_F32):
//   A 16x4 : lanes 0-15 row M=lane (K=0,1); lanes 16-31 row M=lane-16 (K=2,3)
//   B 4x16 : lanes 0-15 col N=lane (K=0,1); lanes 16-31 col N=lane-16 (K=2,3)
//   C 16x16: VGPR r -> M=r (lanes 0-15), M=r+8 (lanes 16-31), N=lane&15
// A-row bounds via clamped address + value mask (v_cndmask, no EXEC
// divergence at the WMMA; EXEC must be all ones).
// ---------------------------------------------------------------------------
template<int CT>
__global__ __launch_bounds__(256) void wmma_gemm_tiled_kernel(
    const float* __restrict__ A, const float* __restrict__ B, float* __restrict__ C,
    int M, int K, int Ncols)
{
    extern __shared__ float W_lds[];          // [K][CT*16], zero padded
    constexpr int NP = CT * 16;

    const int tid = threadIdx.x;

    // --- cooperative staging of B into LDS (zero-pad cols >= Ncols) ---------
    for (int idx = tid; idx < K * NP; idx += 256) {
        const int row = idx / NP;
        const int col = idx % NP;
        W_lds[idx] = (col < Ncols) ? B[(size_t)row * Ncols + col] : 0.0f;
    }
    __syncthreads();

    const int wave  = tid >> 5;
    const int lane  = tid & 31;
    const int tileM = (blockIdx.x * 8 + wave) * 16;

    const int rowA  = tileM + (lane & 15);
    const int rA    = (rowA < M) ? rowA : (M - 1);
    const float mA  = (rowA < M) ? 1.0f : 0.0f;

    const int kOff  = (lane >> 4) << 1;       // 0 for lanes 0-15, 2 for 16-31
    const int colL  = lane & 15;

    v8f acc[CT];
    #pragma unroll
    for (int t = 0; t < CT; ++t) acc[t] = (v8f){};

    for (int k0 = 0; k0 < K; k0 += 4) {
        const float* Ap = A + (size_t)rA * K + (k0 + kOff);
        v2f a;
        a.x = Ap[0] * mA;                      // global_load_b64 + mask
        a.y = Ap[1] * mA;

        const float* Bp = &W_lds[(k0 + kOff) * NP + colL];
        #pragma unroll
        for (int t = 0; t < CT; ++t) {
            v2f b;
            b.x = Bp[t * 16];                  // ds_load
            b.y = Bp[t * 16 + NP];
            acc[t] = __builtin_amdgcn_wmma_f32_16x16x4_f32(
                /*neg_a=*/false, a, /*neg_b=*/false, b,
                /*c_mod=*/(short)0, acc[t], /*reuse_a=*/false, /*reuse_b=*/false);
        }
    }

    const int rowBase = tileM + ((lane >> 4) << 3);   // +8 for upper half-wave
    #pragma unroll
    for (int t = 0; t < CT; ++t) {
        const int colC = t * 16 + colL;
        if (colC < Ncols) {
            #pragma unroll
            for (int r = 0; r < 8; ++r) {
                const int row = rowBase + r;
                if (row < M) C[(size_t)row * Ncols + colC] = acc[t][r];
            }
        }
    }
}

// ---------------------------------------------------------------------------
// Edge scatter: agg[dst[e], d] += h[src[e], d] * dinv[src]*dinv[dst]
// Thread per (edge, feature): consecutive threads hit consecutive features of
// the same row -> coalesced gather reads and coalesced atomics.
// ---------------------------------------------------------------------------
__global__ void scatter_add_kernel(
    const float* __restrict__ h, const float* __restrict__ dinv,
    const int* __restrict__ src, const int* __restrict__ dst,
    float* __restrict__ agg, long long total, int D)
{
    long long tid = (long long)blockIdx.x * blockDim.x + threadIdx.x;
    if (tid >= total) return;
    const int e = (int)(tid / D);
    const int d = (int)(tid % D);
    const int s = src[e];
    const int t = dst[e];
    const float norm = dinv[s] * dinv[t];
    const float v = h[(size_t)s * D + d] * norm;
    atomAddF(&agg[(size_t)t * D + d], v);
}

// ---------------------------------------------------------------------------
// Layer-1 epilogue: h1 = relu(agg1 + h1*dinv^2 + b1) * drop_mask  (in place)
// ---------------------------------------------------------------------------
__global__ void combine1_kernel(
    float* __restrict__ h1, const float* __restrict__ agg1,
    const float* __restrict__ dinv, const float* __restrict__ b1,
    const float* __restrict__ drop, long long total, int D)
{
    long long tid = (long long)blockIdx.x * blockDim.x + threadIdx.x;
    if (tid >= total) return;
    const int i = (int)(tid / D);
    const int d = (int)(tid % D);
    const float di = dinv[i];
    float v = agg1[tid] + h1[tid] * di * di + b1[d];
    v = fmaxf(v, 0.0f) * drop[tid];
    h1[tid] = v;
}

// ---------------------------------------------------------------------------
// Layer-2 epilogue: out += h2*dinv^2 + b2  (out already holds edge-scatter sum)
// ---------------------------------------------------------------------------
__global__ void combine2_kernel(
    float* __restrict__ out, const float* __restrict__ h2,
    const float* __restrict__ dinv, const float* __restrict__ b2,
    long long total, int D)
{
    long long tid = (long long)blockIdx.x * blockDim.x + threadIdx.x;
    if (tid >= total) return;
    const int i = (int)(tid / D);
    const int d = (int)(tid % D);
    const float di = dinv[i];
    out[tid] = out[tid] + h2[tid] * di * di + b2[d];
}

// ---------------------------------------------------------------------------
// Host-side orchestration
// ---------------------------------------------------------------------------
static inline unsigned cdiv_ll(long long a, int b) { return (unsigned)((a + b - 1) / b); }

static void launch_gemm(const float* A, const float* B, float* C,
                        int M, int K, int Ncols, hipStream_t stream) {
    const int ct    = (Ncols + 15) / 16;
    const dim3 grid((M + 127) / 128);
    const size_t sh = (size_t)K * ct * 16 * sizeof(float);
    switch (ct) {
        case 1: wmma_gemm_tiled_kernel<1><<<grid, 256, sh, stream>>>(A, B, C, M, K, Ncols); break;
        case 2: wmma_gemm_tiled_kernel<2><<<grid, 256, sh, stream>>>(A, B, C, M, K, Ncols); break;
        case 3: wmma_gemm_tiled_kernel<3><<<grid, 256, sh, stream>>>(A, B, C, M, K, Ncols); break;
        default:
        case 4: wmma_gemm_tiled_kernel<4><<<grid, 256, sh, stream>>>(A, B, C, M, K, Ncols); break;
    }
}

extern "C" void kernel_launch(void* const* d_in, const int* in_sizes, int n_in,
                              void* d_out, int out_size, void* d_ws, size_t ws_size,
                              hipStream_t stream) {
    const float* x    = (const float*)d_in[0];
    const int*   ei   = (const int*)  d_in[1];
    const float* W1   = (const float*)d_in[2];
    const float* b1   = (const float*)d_in[3];
    const float* W2   = (const float*)d_in[4];
    const float* b2   = (const float*)d_in[5];
    const float* drop = (const float*)d_in[6];

    const int HID  = in_sizes[3];                 // 64
    const int OUT  = in_sizes[5];                 // 40
    const int FEAT = in_sizes[2] / HID;           // 128
    const int N    = in_sizes[0] / FEAT;          // 100000
    const int E    = in_sizes[1] / 2;             // 1600000

    const int* src = ei;
    const int* dst = ei + E;

    // Workspace layout (floats)
    float* ws   = (float*)d_ws;
    float* deg  = ws;                                   // [N]  -> becomes dinv
    float* h1   = deg + (size_t)N;                      // [N, HID]
    float* agg1 = h1  + (size_t)N * HID;                // [N, HID]
    float* h2   = agg1 + (size_t)N * HID;               // [N, OUT]
    float* out  = (float*)d_out;                        // [N, OUT]

    const int BT = 256;

    // 1) init: deg = 1.0 (self-loop), agg1 = 0, out = 0
    fill_kernel<<<cdiv_ll(N, BT), BT, 0, stream>>>(deg, 1.0f, (long long)N);
    fill_kernel<<<cdiv_ll((long long)N * HID, BT), BT, 0, stream>>>(agg1, 0.0f, (long long)N * HID);
    fill_kernel<<<cdiv_ll((long long)N * OUT, BT), BT, 0, stream>>>(out, 0.0f, (long long)N * OUT);

    // 2) degrees, then dinv = rsqrt(deg) in place
    degree_kernel<<<cdiv_ll(E, BT), BT, 0, stream>>>(dst, deg, E);
    rsqrt_kernel<<<cdiv_ll(N, BT), BT, 0, stream>>>(deg, N);
    const float* dinv = deg;

    // 3) GEMM1: h1 = x @ W1   (WMMA fp32, W1 staged in LDS, A read once)
    launch_gemm(x, W1, h1, N, FEAT, HID, stream);

    // 4) edge scatter layer 1
    scatter_add_kernel<<<cdiv_ll((long long)E * HID, BT), BT, 0, stream>>>(
        h1, dinv, src, dst, agg1, (long long)E * HID, HID);

    // 5) epilogue layer 1 (relu + dropout mask), in place into h1
    combine1_kernel<<<cdiv_ll((long long)N * HID, BT), BT, 0, stream>>>(
        h1, agg1, dinv, b1, drop, (long long)N * HID, HID);

    // 6) GEMM2: h2 = h1 @ W2  (OUT=40 zero-padded to 48 in LDS)
    launch_gemm(h1, W2, h2, N, HID, OUT, stream);

    // 7) edge scatter layer 2 directly into out
    scatter_add_kernel<<<cdiv_ll((long long)E * OUT, BT), BT, 0, stream>>>(
        h2, dinv, src, dst, out, (long long)E * OUT, OUT);

    // 8) epilogue layer 2
    combine2_kernel<<<cdiv_ll((long long)N * OUT, BT), BT, 0, stream>>>(
        out, h2, dinv, b2, (long long)N * OUT, OUT);
}